// FeatureRefinementGNN_54030688584380
// MI455X (gfx1250) — compile-verified
//
#include <hip/hip_runtime.h>
#include <hip/hip_bf16.h>
#include <stdint.h>

typedef __attribute__((ext_vector_type(2))) float v2f;
typedef __attribute__((ext_vector_type(8))) float v8f;

#define BN_EPS 1e-5f

// ---------------------------------------------------------------------------
// Degree / normalization kernels
// ---------------------------------------------------------------------------
__global__ void deg_init_kernel(float* __restrict__ deg, int N) {
    int i = blockIdx.x * blockDim.x + threadIdx.x;
    if (i < N) deg[i] = 1.0f;  // self-loop
}

__global__ void deg_count_kernel(const int* __restrict__ col, float* __restrict__ deg, int E) {
    int i = blockIdx.x * blockDim.x + threadIdx.x;
    if (i < E) atomicAdd(&deg[col[i]], 1.0f);
}

__global__ void deg_rsqrt_kernel(float* __restrict__ deg, int N) {
    int i = blockIdx.x * blockDim.x + threadIdx.x;
    if (i < N) deg[i] = rsqrtf(deg[i]);
}

// ---------------------------------------------------------------------------
// FP32 WMMA GEMM: out[M,FOUT] = A[M,128] @ W[128,FOUT]
// FOUT is a compile-time constant so all W/out strides are immediate offsets
// (no per-iteration 64-bit address math in the WMMA loop).
// blockDim.x == 32 * (FOUT/16); one block per 16-row tile of A.
// ---------------------------------------------------------------------------
template <int FOUT>
__global__ void gemm_wmma_kernel(const float* __restrict__ A,
                                 const float* __restrict__ W,
                                 float* __restrict__ out) {
    const int K = 128;
    __shared__ float As[16 * K];  // 8 KB

    const int m0  = blockIdx.x * 16;
    const int tid = threadIdx.x;

    // Cooperative b128 staging of the 16x128 A tile (coalesced).
    const float4* A4  = (const float4*)(A + m0 * K);
    float4*       As4 = (float4*)As;
    for (int i = tid; i < 16 * (K / 4); i += blockDim.x) As4[i] = A4[i];
    __syncthreads();

    const int wave = tid >> 5;
    const int lane = tid & 31;
    const int half = lane >> 4;   // 0: lanes 0-15, 1: lanes 16-31
    const int l    = lane & 15;
    const int n0   = wave * 16;

    // Pointer walks with compile-time strides.
    const float* Ap = As + l * K + 2 * half;              // A frag: K={k,k+1}|{k+2,k+3}
    const float* Wp = W + (2 * half) * FOUT + n0 + l;     // B frag: same K split, col n0+l

    v8f c = {};
    #pragma unroll
    for (int k = 0; k < K; k += 4) {
        v2f a, b;
        a.x = Ap[0];
        a.y = Ap[1];
        b.x = Wp[0];
        b.y = Wp[FOUT];
        Ap += 4;
        Wp += 4 * FOUT;
        c = __builtin_amdgcn_wmma_f32_16x16x4_f32(false, a, false, b,
                                                  (short)0, c, false, false);
    }

    // C/D layout: VGPR j -> M = j (lanes 0-15) / j+8 (lanes 16-31), N = l
    float* Op = out + (m0 + 8 * half) * FOUT + n0 + l;
    #pragma unroll
    for (int j = 0; j < 8; ++j) Op[j * FOUT] = c[j];
}

// ---------------------------------------------------------------------------
// GCN propagate: out = selfloop + scatter. Bias folded into the init pass
// (equivalent to adding b after segment_sum).
// ---------------------------------------------------------------------------
__global__ void selfloop_init_kernel(const float* __restrict__ t,
                                     const float* __restrict__ dinv,
                                     const float* __restrict__ bias,
                                     float* __restrict__ out,
                                     int total, int Fshift) {
    int idx = blockIdx.x * blockDim.x + threadIdx.x;
    if (idx >= total) return;
    int f = idx & ((1 << Fshift) - 1);
    int i = idx >> Fshift;
    float d = dinv[i];
    out[idx] = t[idx] * d * d + bias[f];
}

__global__ void edge_scatter_kernel(const float* __restrict__ t,
                                    const int* __restrict__ row,
                                    const int* __restrict__ col,
                                    const float* __restrict__ dinv,
                                    float* __restrict__ out,
                                    int total, int Fshift) {
    int idx = blockIdx.x * blockDim.x + threadIdx.x;
    if (idx >= total) return;
    int f = idx & ((1 << Fshift) - 1);
    int e = idx >> Fshift;
    int r = row[e];
    int c = col[e];
    float norm = dinv[r] * dinv[c];
    atomicAdd(&out[(c << Fshift) + f], t[(r << Fshift) + f] * norm);
}

// ---------------------------------------------------------------------------
// BatchNorm (F = 128 fixed): stats then apply+ReLU (in place)
// ---------------------------------------------------------------------------
__global__ void zero_sums_kernel(float* __restrict__ sum, float* __restrict__ sumsq) {
    int t = threadIdx.x;
    if (t < 128) { sum[t] = 0.0f; sumsq[t] = 0.0f; }
}

__global__ void bn_stats_kernel(const float* __restrict__ x,
                                float* __restrict__ sum,
                                float* __restrict__ sumsq, int N) {
    const int F = 128;
    const int ROWS = 2048;
    int f  = threadIdx.x & (F - 1);
    int rh = threadIdx.x >> 7;   // 0 or 1
    int base = blockIdx.x * ROWS;
    int end  = min(N, base + ROWS);
    float s = 0.0f, s2 = 0.0f;
    for (int r = base + rh; r < end; r += 2) {   // coalesced: lanes span features
        float v = x[r * F + f];
        s += v; s2 += v * v;
    }
    __shared__ float red[256];
    red[threadIdx.x] = s;
    __syncthreads();
    if (rh == 0) atomicAdd(&sum[f], red[f] + red[f + 128]);
    __syncthreads();
    red[threadIdx.x] = s2;
    __syncthreads();
    if (rh == 0) atomicAdd(&sumsq[f], red[f] + red[f + 128]);
}

__global__ void bn_apply_kernel(float* __restrict__ x,
                                const float* __restrict__ sum,
                                const float* __restrict__ sumsq,
                                const float* __restrict__ g,
                                const float* __restrict__ beta, int N) {
    const int F = 128;
    int idx = blockIdx.x * blockDim.x + threadIdx.x;
    if (idx >= N * F) return;
    int f = idx & (F - 1);
    float invN = 1.0f / (float)N;
    float mu  = sum[f] * invN;
    float var = sumsq[f] * invN - mu * mu;
    float inv = rsqrtf(var + BN_EPS);
    float v = (x[idx] - mu) * inv * g[f] + beta[f];
    x[idx] = fmaxf(v, 0.0f);
}

// ---------------------------------------------------------------------------
// Edge MLP: ew[e] = sigmoid(relu(concat(refined[row],refined[col]) @ eW1 + eb1) @ eW2 + eb2)
// Block = 16 edges, 256 threads (8 waves x 16 hidden cols each), WMMA hidden GEMM.
// ---------------------------------------------------------------------------
__global__ void edge_mlp_kernel(const float* __restrict__ refined,
                                const int* __restrict__ row,
                                const int* __restrict__ col,
                                const float* __restrict__ eW1,
                                const float* __restrict__ eb1,
                                const float* __restrict__ eW2,
                                const float* __restrict__ eb2,
                                float* __restrict__ ew, int E) {
    const int K = 128;     // 2*DOUT edge-feature width == hidden width
    __shared__ float As[16 * K];   // gathered edge features, 8 KB
    __shared__ float partial[16];

    int e0  = blockIdx.x * 16;
    int tid = threadIdx.x;

    // Gather concat(refined[row[e]], refined[col[e]]) for 16 edges (b128 loads:
    // each refined row is 16 aligned float4s).
    const float4* R4  = (const float4*)refined;
    float4*       As4 = (float4*)As;
    for (int i = tid; i < 16 * (K / 4); i += blockDim.x) {
        int m = i >> 5;           // edge slot 0..15
        int q = i & 31;           // float4 index within 128-float row
        int e = min(e0 + m, E - 1);
        int node = (q < 16) ? row[e] : col[e];
        As4[i] = R4[node * 16 + (q & 15)];
    }
    if (tid < 16) partial[tid] = 0.0f;
    __syncthreads();

    const int wave = tid >> 5;
    const int lane = tid & 31;
    const int half = lane >> 4;
    const int l    = lane & 15;
    const int n0   = wave * 16;

    const float* Ap = As + l * K + 2 * half;
    const float* Wp = eW1 + (2 * half) * 128 + n0 + l;

    v8f c = {};
    #pragma unroll
    for (int k = 0; k < K; k += 4) {
        v2f a, b;
        a.x = Ap[0];
        a.y = Ap[1];
        b.x = Wp[0];
        b.y = Wp[128];
        Ap += 4;
        Wp += 4 * 128;
        c = __builtin_amdgcn_wmma_f32_16x16x4_f32(false, a, false, b,
                                                  (short)0, c, false, false);
    }

    float bhid = eb1[n0 + l];
    float w2   = eW2[n0 + l];
    #pragma unroll
    for (int j = 0; j < 8; ++j) {
        float h = fmaxf(c[j] + bhid, 0.0f);          // ReLU(hidden)
        atomicAdd(&partial[j + 8 * half], h * w2);   // ds_add_f32 reduction
    }
    __syncthreads();

    if (tid < 16) {
        int e = e0 + tid;
        if (e < E) {
            float v = partial[tid] + eb2[0];
            ew[e] = 1.0f / (1.0f + expf(-v));
        }
    }
}

// ---------------------------------------------------------------------------
// Host-side orchestration
// ---------------------------------------------------------------------------
static inline int cdiv(long long a, long long b) { return (int)((a + b - 1) / b); }

extern "C" void kernel_launch(void* const* d_in, const int* in_sizes, int n_in,
                              void* d_out, int out_size, void* d_ws, size_t ws_size,
                              hipStream_t stream) {
    const float* x    = (const float*)d_in[0];
    const int*   ei   = (const int*)  d_in[1];
    const float* W1   = (const float*)d_in[2];
    const float* b1   = (const float*)d_in[3];
    const float* W2   = (const float*)d_in[4];
    const float* b2   = (const float*)d_in[5];
    const float* W3   = (const float*)d_in[6];
    const float* b3   = (const float*)d_in[7];
    const float* g1   = (const float*)d_in[8];
    const float* bt1  = (const float*)d_in[9];
    const float* g2   = (const float*)d_in[10];
    const float* bt2  = (const float*)d_in[11];
    const float* eW1  = (const float*)d_in[12];
    const float* eb1  = (const float*)d_in[13];
    const float* eW2  = (const float*)d_in[14];
    const float* eb2  = (const float*)d_in[15];

    const int N = in_sizes[0] / 128;   // 100000
    const int E = in_sizes[1] / 2;     // 1600000
    const int* row = ei;
    const int* col = ei + E;

    float* refined = (float*)d_out;                 // [N, 64]
    float* ew      = refined + (size_t)N * 64;      // [E]

    // Workspace carve-up (~103 MB)
    float* dinv  = (float*)d_ws;                    // N
    float* bufA  = dinv + (size_t)N;                // N*128
    float* bufB  = bufA + (size_t)N * 128;          // N*128
    float* sum   = bufB + (size_t)N * 128;          // 128
    float* sumsq = sum + 128;                       // 128

    const int BS = 256;
    const int nNF128 = N * 128;
    const int nNF64  = N * 64;
    const int nEF128 = E * 128;    // 204.8M, fits int32
    const int nEF64  = E * 64;

    // --- degree / D^-1/2 ---
    deg_init_kernel <<<cdiv(N, BS), BS, 0, stream>>>(dinv, N);
    deg_count_kernel<<<cdiv(E, BS), BS, 0, stream>>>(col, dinv, E);
    deg_rsqrt_kernel<<<cdiv(N, BS), BS, 0, stream>>>(dinv, N);

    int bnBlocks = cdiv(N, 2048);

    // --- layer 1: GCNConv(x, W1) -> BN -> ReLU ---
    gemm_wmma_kernel<128><<<N / 16, 256, 0, stream>>>(x, W1, bufA);
    selfloop_init_kernel<<<cdiv(nNF128, BS), BS, 0, stream>>>(bufA, dinv, b1, bufB, nNF128, 7);
    edge_scatter_kernel <<<cdiv(nEF128, BS), BS, 0, stream>>>(bufA, row, col, dinv, bufB, nEF128, 7);
    zero_sums_kernel<<<1, 256, 0, stream>>>(sum, sumsq);
    bn_stats_kernel <<<bnBlocks, 256, 0, stream>>>(bufB, sum, sumsq, N);
    bn_apply_kernel <<<cdiv(nNF128, BS), BS, 0, stream>>>(bufB, sum, sumsq, g1, bt1, N);

    // --- layer 2: GCNConv(h1, W2) -> BN -> ReLU ---
    gemm_wmma_kernel<128><<<N / 16, 256, 0, stream>>>(bufB, W2, bufA);
    selfloop_init_kernel<<<cdiv(nNF128, BS), BS, 0, stream>>>(bufA, dinv, b2, bufB, nNF128, 7);
    edge_scatter_kernel <<<cdiv(nEF128, BS), BS, 0, stream>>>(bufA, row, col, dinv, bufB, nEF128, 7);
    zero_sums_kernel<<<1, 256, 0, stream>>>(sum, sumsq);
    bn_stats_kernel <<<bnBlocks, 256, 0, stream>>>(bufB, sum, sumsq, N);
    bn_apply_kernel <<<cdiv(nNF128, BS), BS, 0, stream>>>(bufB, sum, sumsq, g2, bt2, N);

    // --- layer 3: GCNConv(h2, W3) -> refined (written straight to d_out) ---
    gemm_wmma_kernel<64><<<N / 16, 128, 0, stream>>>(bufB, W3, bufA);
    selfloop_init_kernel<<<cdiv(nNF64, BS), BS, 0, stream>>>(bufA, dinv, b3, refined, nNF64, 6);
    edge_scatter_kernel <<<cdiv(nEF64, BS), BS, 0, stream>>>(bufA, row, col, dinv, refined, nEF64, 6);

    // --- edge MLP + sigmoid ---
    edge_mlp_kernel<<<cdiv(E, 16), 256, 0, stream>>>(refined, row, col,
                                                     eW1, eb1, eW2, eb2, ew, E);
}